// CTRLightGCN_26139170963557
// MI455X (gfx1250) — compile-verified
//
#include <hip/hip_runtime.h>

// ---------------- problem constants ----------------
#define Bn   64
#define Cn   256
#define Tn   128
#define Vn   25
#define Gn   4
#define Cg   64          // channels per group
#define TVn  3200        // T*V
#define TC   16          // t-chunk per workgroup
#define NC   400         // TC*Vn columns per chunk (25 N-tiles of 16)
#define XS_STRIDE 404    // padded LDS row stride for x chunk (mult of 4, bank-spread)
#define WS_STRIDE 68     // padded LDS row stride for conv_w (mult of 4, bank-spread)
#define AG_STRIDE 26     // padded A_g row stride (even -> 8B-aligned w-pairs)
#define NTOT 204800.0f   // B*T*V elements per channel for BN stats

// ---------------- workspace layout (floats) ----------------
#define WS_AG    0       // G*V*V = 2500
#define WS_SUM   2500    // 256
#define WS_SQ    2756    // 256
#define WS_SCALE 3012    // 256
#define WS_SHIFT 3268    // 256

typedef __attribute__((ext_vector_type(2))) float v2f;
typedef __attribute__((ext_vector_type(8))) float v8f;

// =====================================================================
// Kernel 1: build A_g = A_physical + A_group, zero stat accumulators.
// =====================================================================
__global__ void __launch_bounds__(512) gcn_prep_kernel(const float* __restrict__ A,
                                                       const float* __restrict__ Agrp,
                                                       float* __restrict__ w) {
  int tid = threadIdx.x;
  if (tid < 512) w[WS_SUM + tid] = 0.0f;   // zero sum(256) + sumsq(256)
  if (tid < Vn) {
    float row[Vn];
    float rs = 0.0f;
    #pragma unroll
    for (int j = 0; j < Vn; j++) {
      float s = A[0 * Vn * Vn + tid * Vn + j]
              + A[1 * Vn * Vn + tid * Vn + j]
              + A[2 * Vn * Vn + tid * Vn + j];
      row[j] = s;
      rs += s;
    }
    rs = fmaxf(rs, 1e-6f);
    float inv = 1.0f / rs;
    for (int g = 0; g < Gn; g++)
      #pragma unroll
      for (int j = 0; j < Vn; j++)
        w[WS_AG + (g * Vn + tid) * Vn + j] = row[j] * inv + Agrp[(g * Vn + tid) * Vn + j];
  }
}

// =====================================================================
// Shared device helpers for the two GEMM passes
// =====================================================================
// Stage the x chunk with CDNA5 async global->LDS B128 copies (ASYNCcnt path),
// stage conv_w + A_g (padded stride 26) with ordinary loads.
__device__ __forceinline__ void load_stage(const float* __restrict__ x,
                                           const float* __restrict__ convw,
                                           const float* __restrict__ wAg,
                                           float* xs, float* Wsm, float* sAgP,
                                           int b, int g, int t0) {
  int tid = threadIdx.x;
  // x chunk: 64 rows (c) x 400 cols (t_local*25+v), 16B async copies to LDS
  const float* xbase = x + ((size_t)(b * Cn + g * Cg)) * TVn + (size_t)t0 * Vn;
  for (int i = tid; i < 64 * 100; i += 256) {
    int row = i / 100, j = i - row * 100;
    const float* gp = xbase + (size_t)row * TVn + 4 * j;
    unsigned lds_addr = (unsigned)(size_t)(&xs[row * XS_STRIDE + 4 * j]);
    unsigned long long gaddr = (unsigned long long)(size_t)gp;
    asm volatile("global_load_async_to_lds_b128 %0, %1, off"
                 :: "v"(lds_addr), "v"(gaddr) : "memory");
  }
  // conv_w for this group: 64x64 -> padded 64x68
  const float4* wsrc = (const float4*)(convw + g * Cg * Cg);
  for (int i = tid; i < 1024; i += 256) {
    int row = i >> 4, col = (i & 15) * 4;
    *(float4*)(&Wsm[row * WS_STRIDE + col]) = wsrc[i];
  }
  // A_g for this group, padded to stride 26 so w-pairs are 8B aligned
  for (int i = tid; i < Vn * Vn; i += 256) {
    int v = i / 25, ww = i - v * 25;
    sAgP[v * AG_STRIDE + ww] = wAg[g * Vn * Vn + i];
  }
  // drain the async copies issued by this wave before the workgroup barrier
  asm volatile("s_wait_asynccnt 0x0" ::: "memory");
}

// In-place spatial aggregation in LDS: xs[c, t*25+v] = sum_w Ag[v,w]*xs[c, t*25+w]
// Packed f32 FMA (v_pk_fma_f32) over w-pairs; each A_g row-pair ds_load_b64 is
// reused across the 4 (c,t) rows this thread owns.
__device__ __forceinline__ void aggregate(float* xs, const float* sAgP) {
  int tid = threadIdx.x;
  float xr[4][Vn];                     // 4 rows x 25 inputs, register resident
  float* p[4];
  #pragma unroll
  for (int k = 0; k < 4; k++) {
    int task = tid + k * 256;          // 1024 tasks = 64 c * 16 t
    int c = task & 63, t = task >> 6;
    p[k] = &xs[c * XS_STRIDE + t * Vn];
    #pragma unroll
    for (int ww = 0; ww < Vn; ww++) xr[k][ww] = p[k][ww];
  }
  #pragma unroll
  for (int v = 0; v < Vn; v++) {
    v2f acc[4];
    #pragma unroll
    for (int k = 0; k < 4; k++) acc[k] = (v2f){0.f, 0.f};
    #pragma unroll
    for (int wp = 0; wp < 12; wp++) {
      v2f a = *(const v2f*)&sAgP[v * AG_STRIDE + 2 * wp];   // 8B-aligned pair
      #pragma unroll
      for (int k = 0; k < 4; k++) {
        v2f xv;
        xv.x = xr[k][2 * wp];
        xv.y = xr[k][2 * wp + 1];
        acc[k] = __builtin_elementwise_fma(xv, a, acc[k]);  // v_pk_fma_f32
      }
    }
    float tail = sAgP[v * AG_STRIDE + 24];
    #pragma unroll
    for (int k = 0; k < 4; k++)
      p[k][v] = acc[k].x + acc[k].y + tail * xr[k][24];
  }
}

// Preload ALL A-fragments (conv_w) for this wave's 4 M-tiles x 16 K-steps
// into registers: 16x4 v2f = 128 VGPRs, reused across every N-tile.
__device__ __forceinline__ void preload_a(const float* Wsm, int lane, v2f afr[16][4]) {
  int half = lane >> 4, lm = lane & 15;
  #pragma unroll
  for (int ks = 0; ks < 16; ks++) {
    int ka = ks * 4 + 2 * half;        // lanes 0-15: K={kk,kk+1}; 16-31: K={kk+2,kk+3}
    #pragma unroll
    for (int mi = 0; mi < 4; mi++)
      afr[ks][mi] = *(const v2f*)&Wsm[(mi * 16 + lm) * WS_STRIDE + ka];
  }
}

// All 4 16x16 M-tiles of one N-column-tile: 64 WMMAs, only 2 LDS loads per 4 WMMAs.
__device__ __forceinline__ void compute_col_tiles(const float* xs, const v2f afr[16][4],
                                                  int n0, int lane, v8f acc[4]) {
  int half = lane >> 4, lm = lane & 15;
  #pragma unroll
  for (int ks = 0; ks < 16; ks++) {
    int ka = ks * 4 + 2 * half;
    v2f bf;
    bf.x = xs[ka * XS_STRIDE + n0 + lm];
    bf.y = xs[(ka + 1) * XS_STRIDE + n0 + lm];
    #pragma unroll
    for (int mi = 0; mi < 4; mi++)
      acc[mi] = __builtin_amdgcn_wmma_f32_16x16x4_f32(false, afr[ks][mi], false, bf,
                                                      (short)0, acc[mi], false, false);
  }
}

// =====================================================================
// Kernel 2: stats pass — compute z tiles, accumulate per-channel sum/sumsq
// =====================================================================
__global__ void __launch_bounds__(256) gcn_stats_kernel(const float* __restrict__ x,
                                                        const float* __restrict__ convw,
                                                        float* __restrict__ w) {
  extern __shared__ float smem[];
  float* xs   = smem;                      // 64*404
  float* Wsm  = xs + 64 * XS_STRIDE;       // 64*68
  float* sAgP = Wsm + 64 * WS_STRIDE;      // 25*26
  float* red  = sAgP + Vn * AG_STRIDE;     // 128: [0..63]=sum, [64..127]=sumsq

  int b = blockIdx.z, g = blockIdx.y, t0 = blockIdx.x * TC;
  int tid = threadIdx.x;
  if (tid < 128) red[tid] = 0.0f;
  load_stage(x, convw, w + WS_AG, xs, Wsm, sAgP, b, g, t0);
  __syncthreads();

  int wave = tid >> 5, lane = tid & 31;
  int half = lane >> 4, lm = lane & 15;
  aggregate(xs, sAgP);
  v2f afr[16][4];
  preload_a(Wsm, lane, afr);               // after aggregate: keeps VGPR peak low
  __syncthreads();

  for (int ni = wave; ni < 25; ni += 8) {
    int n0 = ni * 16;
    v8f acc[4];
    #pragma unroll
    for (int mi = 0; mi < 4; mi++) acc[mi] = (v8f){0.f,0.f,0.f,0.f,0.f,0.f,0.f,0.f};
    compute_col_tiles(xs, afr, n0, lane, acc);
    #pragma unroll
    for (int mi = 0; mi < 4; mi++) {
      float s[8], q[8];
      #pragma unroll
      for (int r = 0; r < 8; r++) { s[r] = acc[mi][r]; q[r] = acc[mi][r] * acc[mi][r]; }
      #pragma unroll
      for (int m = 1; m < 16; m <<= 1) {
        #pragma unroll
        for (int r = 0; r < 8; r++) {
          s[r] += __shfl_xor(s[r], m, 32);
          q[r] += __shfl_xor(q[r], m, 32);
        }
      }
      if (lm == 0) {
        #pragma unroll
        for (int r = 0; r < 8; r++) {
          int row = mi * 16 + r + 8 * half;
          atomicAdd(&red[row], s[r]);
          atomicAdd(&red[64 + row], q[r]);
        }
      }
    }
  }
  __syncthreads();
  if (tid < 64)        atomicAdd(&w[WS_SUM + g * Cg + tid], red[tid]);
  else if (tid < 128)  atomicAdd(&w[WS_SQ  + g * Cg + (tid - 64)], red[tid]);
}

// =====================================================================
// Kernel 3: finalize BN stats -> per-channel scale/shift
// =====================================================================
__global__ void __launch_bounds__(256) gcn_finalize_kernel(const float* __restrict__ gamma,
                                                           const float* __restrict__ beta,
                                                           float* __restrict__ w) {
  int c = threadIdx.x;
  const float invN = 1.0f / NTOT;
  float mean = w[WS_SUM + c] * invN;
  float var  = w[WS_SQ + c] * invN - mean * mean;
  float inv  = rsqrtf(var + 1e-5f);
  float sc   = gamma[c] * inv;
  w[WS_SCALE + c] = sc;
  w[WS_SHIFT + c] = beta[c] - mean * sc;
}

// =====================================================================
// Kernel 4: apply pass — recompute z tiles, out = x + z*scale + shift
// =====================================================================
__global__ void __launch_bounds__(256) gcn_apply_kernel(const float* __restrict__ x,
                                                        const float* __restrict__ convw,
                                                        const float* __restrict__ w,
                                                        float* __restrict__ out) {
  extern __shared__ float smem[];
  float* xs   = smem;
  float* Wsm  = xs + 64 * XS_STRIDE;
  float* sAgP = Wsm + 64 * WS_STRIDE;
  float* red  = sAgP + Vn * AG_STRIDE;     // [0..63]=scale, [64..127]=shift for this group

  int b = blockIdx.z, g = blockIdx.y, t0 = blockIdx.x * TC;
  int tid = threadIdx.x;
  if (tid < 64)       red[tid] = w[WS_SCALE + g * Cg + tid];
  else if (tid < 128) red[tid] = w[WS_SHIFT + g * Cg + (tid - 64)];
  load_stage(x, convw, w + WS_AG, xs, Wsm, sAgP, b, g, t0);
  __syncthreads();

  int wave = tid >> 5, lane = tid & 31;
  int half = lane >> 4, lm = lane & 15;
  aggregate(xs, sAgP);
  v2f afr[16][4];
  preload_a(Wsm, lane, afr);
  __syncthreads();

  size_t chunkoff = (size_t)t0 * Vn;
  for (int ni = wave; ni < 25; ni += 8) {
    int n0 = ni * 16;
    v8f acc[4];
    #pragma unroll
    for (int mi = 0; mi < 4; mi++) acc[mi] = (v8f){0.f,0.f,0.f,0.f,0.f,0.f,0.f,0.f};
    compute_col_tiles(xs, afr, n0, lane, acc);
    #pragma unroll
    for (int mi = 0; mi < 4; mi++) {
      #pragma unroll
      for (int r = 0; r < 8; r++) {
        int row = mi * 16 + r + 8 * half;
        int ch  = g * Cg + row;
        size_t off = ((size_t)(b * Cn + ch)) * TVn + chunkoff + (size_t)(n0 + lm);
        out[off] = x[off] + acc[mi][r] * red[row] + red[64 + row];
      }
    }
  }
}

// =====================================================================
// Host launcher
// =====================================================================
extern "C" void kernel_launch(void* const* d_in, const int* in_sizes, int n_in,
                              void* d_out, int out_size, void* d_ws, size_t ws_size,
                              hipStream_t stream) {
  const float* x     = (const float*)d_in[0];
  const float* A     = (const float*)d_in[1];
  const float* Agrp  = (const float*)d_in[2];
  const float* convw = (const float*)d_in[3];
  const float* gamma = (const float*)d_in[4];
  const float* beta  = (const float*)d_in[5];
  float* out = (float*)d_out;
  float* w   = (float*)d_ws;

  size_t smem = (size_t)(64 * XS_STRIDE + 64 * WS_STRIDE + Vn * AG_STRIDE + 128) * sizeof(float);
  dim3 grid(Tn / TC, Gn, Bn);   // 8 x 4 x 64 = 2048 workgroups

  gcn_prep_kernel<<<1, 512, 0, stream>>>(A, Agrp, w);
  gcn_stats_kernel<<<grid, 256, smem, stream>>>(x, convw, w);
  gcn_finalize_kernel<<<1, 256, 0, stream>>>(gamma, beta, w);
  gcn_apply_kernel<<<grid, 256, smem, stream>>>(x, convw, w, out);
}